// PVAM_11965778886961
// MI455X (gfx1250) — compile-verified
//
#include <hip/hip_runtime.h>
#include <math.h>

typedef __attribute__((ext_vector_type(2))) float v2f;
typedef __attribute__((ext_vector_type(8))) float v8f;

static constexpr int Bb = 32, Tt = 256, Cc = 512, Ll = 25;

__device__ __forceinline__ float fast_tanh(float x) {
#if __has_builtin(__builtin_amdgcn_tanhf)
  return __builtin_amdgcn_tanhf(x);          // gfx1250 TRANS v_tanh_f32
#elif __has_builtin(__builtin_amdgcn_tanh_f32)
  return __builtin_amdgcn_tanh_f32(x);
#else
  return tanhf(x);
#endif
}

// -------------------------------------------------------------------------
// Kernel 1: wf[b*T+t][d] = sum_c X[b*T+t][c] * W[d][c] + bias[d]
// M=8192, N=512, K=512 fp32 GEMM via V_WMMA_F32_16X16X4_F32.
// One wave computes a 64x32 output tile (4x2 grid of 16x16 accumulators):
// 6 b64 fragment loads feed 8 WMMAs per K-step (0.75 loads/WMMA).
// -------------------------------------------------------------------------
__global__ __launch_bounds__(256) void k_gemm_wf(
    const float* __restrict__ X, const float* __restrict__ W,
    const float* __restrict__ bias, float* __restrict__ wf) {
  const int wave = (blockIdx.x * blockDim.x + threadIdx.x) >> 5;  // 0..2047
  const int lane = threadIdx.x & 31;
  const int tileM = wave >> 4;   // 128 M-tiles of 64
  const int tileN = wave & 15;   // 16 N-tiles of 32
  const int m0 = tileM * 64, n0 = tileN * 32;
  const int l = lane & 15, h = lane >> 4;   // A: M=l, K-pair by half; B: N=l
  const int kh = h * 2;

  const float* xr[4];
#pragma unroll
  for (int mi = 0; mi < 4; ++mi)
    xr[mi] = X + (size_t)(m0 + mi * 16 + l) * Cc + kh;
  const float* wr[2];
#pragma unroll
  for (int ni = 0; ni < 2; ++ni)
    wr[ni] = W + (size_t)(n0 + ni * 16 + l) * Cc + kh;

  v8f acc[4][2] = {};
  for (int k = 0; k < Cc; k += 4) {
    v2f a[4], bf[2];
#pragma unroll
    for (int mi = 0; mi < 4; ++mi) a[mi] = *(const v2f*)(xr[mi] + k);
#pragma unroll
    for (int ni = 0; ni < 2; ++ni) bf[ni] = *(const v2f*)(wr[ni] + k);
#pragma unroll
    for (int mi = 0; mi < 4; ++mi)
#pragma unroll
      for (int ni = 0; ni < 2; ++ni)
        acc[mi][ni] = __builtin_amdgcn_wmma_f32_16x16x4_f32(
            false, a[mi], false, bf[ni], (short)0, acc[mi][ni], false, false);
  }

  const float bn[2] = {bias[n0 + l], bias[n0 + 16 + l]};
#pragma unroll
  for (int mi = 0; mi < 4; ++mi) {
#pragma unroll
    for (int r = 0; r < 8; ++r) {
      const int row = m0 + mi * 16 + r + h * 8;   // C/D layout: M = r + 8*half
#pragma unroll
      for (int ni = 0; ni < 2; ++ni)
        wf[(size_t)row * Cc + n0 + ni * 16 + l] = acc[mi][ni][r] + bn[ni];
    }
  }
}

// -------------------------------------------------------------------------
// Kernel 2: scores[b][l][t] = sum_c tanh(wf[b,t,c] + pos[l,c]) * aw[c] + ab
// One wave per (b,t); wf row + atten_w live in registers, reused for all 25 l.
// -------------------------------------------------------------------------
__global__ __launch_bounds__(256) void k_scores(
    const float* __restrict__ wf, const float* __restrict__ pos,
    const float* __restrict__ aw, const float* __restrict__ ab,
    float* __restrict__ scores) {
  const int wave = (blockIdx.x * blockDim.x + threadIdx.x) >> 5;  // 0..8191
  const int lane = threadIdx.x & 31;
  const int b = wave / Tt, t = wave % Tt;

  const float* wrow = wf + (size_t)(b * Tt + t) * Cc;
  float wreg[16], areg[16];
#pragma unroll
  for (int j = 0; j < 16; ++j) {
    wreg[j] = wrow[lane + 32 * j];
    areg[j] = aw[lane + 32 * j];
  }
  const float abv = ab[0];

  for (int li = 0; li < Ll; ++li) {
    const float* prow = pos + li * Cc;
    float acc = 0.f;
#pragma unroll
    for (int j = 0; j < 16; ++j)
      acc += fast_tanh(wreg[j] + prow[lane + 32 * j]) * areg[j];
#pragma unroll
    for (int off = 16; off > 0; off >>= 1)
      acc += __shfl_xor(acc, off, 32);
    if (lane == 0)
      scores[((size_t)b * Ll + li) * Tt + t] = acc + abv;
  }
}

// -------------------------------------------------------------------------
// Kernel 3: softmax over t (256) for each of the 800 (b,l) rows; 1 wave/row.
// -------------------------------------------------------------------------
__global__ __launch_bounds__(256) void k_softmax(float* __restrict__ scores) {
  const int wave = (blockIdx.x * blockDim.x + threadIdx.x) >> 5;  // 0..799
  const int lane = threadIdx.x & 31;
  float* row = scores + (size_t)wave * Tt;

  float v[8];
#pragma unroll
  for (int j = 0; j < 8; ++j) v[j] = row[lane + 32 * j];
  float m = v[0];
#pragma unroll
  for (int j = 1; j < 8; ++j) m = fmaxf(m, v[j]);
#pragma unroll
  for (int off = 16; off > 0; off >>= 1)
    m = fmaxf(m, __shfl_xor(m, off, 32));
  float s = 0.f;
#pragma unroll
  for (int j = 0; j < 8; ++j) { v[j] = __expf(v[j] - m); s += v[j]; }
#pragma unroll
  for (int off = 16; off > 0; off >>= 1)
    s += __shfl_xor(s, off, 32);
  const float inv = 1.0f / s;
#pragma unroll
  for (int j = 0; j < 8; ++j) row[lane + 32 * j] = v[j] * inv;
}

// -------------------------------------------------------------------------
// Kernel 4: out[b][m][n] = sum_t attn[b][m][t] * wf[b][t][n]
// Per batch: (25 pad 32) x 256 x 512 GEMM via fp32 WMMA. M padded with a
// multiplicative zero mask (loads stay in-bounds; EXEC stays all-ones).
// One wave per (b, mtile, ntile): 32*2*32 = 2048 waves.
// -------------------------------------------------------------------------
__global__ __launch_bounds__(256) void k_bmm(
    const float* __restrict__ attn, const float* __restrict__ wf,
    float* __restrict__ out) {
  const int wave = (blockIdx.x * blockDim.x + threadIdx.x) >> 5;  // 0..2047
  const int lane = threadIdx.x & 31;
  const int b = wave >> 6;
  const int rem = wave & 63;
  const int mt = rem >> 5;         // 0..1
  const int nt = rem & 31;         // 0..31
  const int m0 = mt * 16, n0 = nt * 16;
  const int l = lane & 15, h = lane >> 4;
  const int kh = h * 2;

  const int ra = m0 + l;
  const float amask = (ra < Ll) ? 1.f : 0.f;
  const float* arow = attn + ((size_t)b * Ll + (ra < Ll ? ra : 0)) * Tt;
  const float* wbase = wf + ((size_t)b * Tt) * Cc + n0 + l;

  v8f c = {};
  for (int k = 0; k < Tt; k += 4) {
    v2f a = *(const v2f*)(arow + k + kh);
    a.x *= amask;
    a.y *= amask;
    v2f bf;
    bf.x = wbase[(size_t)(k + kh) * Cc];
    bf.y = wbase[(size_t)(k + kh + 1) * Cc];
    c = __builtin_amdgcn_wmma_f32_16x16x4_f32(false, a, false, bf, (short)0, c, false, false);
  }

#pragma unroll
  for (int r = 0; r < 8; ++r) {
    const int m = m0 + r + h * 8;
    if (m < Ll)
      out[((size_t)b * Ll + m) * Cc + n0 + l] = c[r];
  }
}

// -------------------------------------------------------------------------
extern "C" void kernel_launch(void* const* d_in, const int* in_sizes, int n_in,
                              void* d_out, int out_size, void* d_ws, size_t ws_size,
                              hipStream_t stream) {
  const float* X    = (const float*)d_in[0];   // word_features (32,256,512)
  const float* W    = (const float*)d_in[1];   // word_fc_w (512,512)
  const float* bias = (const float*)d_in[2];   // word_fc_b (512)
  const float* pos  = (const float*)d_in[3];   // pos_emb (25,512)
  const float* aw   = (const float*)d_in[4];   // atten_w (512)
  const float* ab   = (const float*)d_in[5];   // atten_b (1)
  float* out = (float*)d_out;                  // (32,25,512)

  float* wf     = (float*)d_ws;                         // 8192*512 f32 = 16 MB
  float* scores = wf + (size_t)Bb * Tt * Cc;            // 32*25*256 f32

  k_gemm_wf<<<256, 256, 0, stream>>>(X, W, bias, wf);           // 2048 waves
  k_scores <<<1024, 256, 0, stream>>>(wf, pos, aw, ab, scores); // 8192 waves
  k_softmax<<<100, 256, 0, stream>>>(scores);                   // 800 waves
  k_bmm    <<<256, 256, 0, stream>>>(scores, wf, out);          // 2048 waves
}